// Matcher_88373247082687
// MI455X (gfx1250) — compile-verified
//
#include <hip/hip_runtime.h>
#include <hip/hip_bf16.h>
#include <math.h>
#include <stdint.h>

// Problem constants (from the reference)
#define B_    128
#define TQ_   64
#define F_    512
#define D_    256
#define C_    100
#define CP_   112          // C padded to multiple of 16 for WMMA tiling
#define KSEL_ 10
#define N_    (B_*TQ_)     // 8192
#define H_    128
#define TEMP_ 0.1f
#define TAU_  1.0f

// GEMM blocking
#define MBLK_ 128          // 8 waves x 16 rows
#define NBLK_ 64           // 4 x 16-col tiles per wave
#define KB_   32           // K panel per stage
#define LDA_  36           // sA row stride (floats): b64 reads conflict-free
#define LDB_  80           // sB row stride (floats): halves hit disjoint bank groups

typedef float v2f __attribute__((ext_vector_type(2)));
typedef float v8f __attribute__((ext_vector_type(8)));

// V_WMMA_F32_16X16X4_F32: A=16x4 f32 (2 VGPRs), B=4x16 f32 (2 VGPRs), C/D=16x16 f32 (8 VGPRs)
__device__ __forceinline__ v8f wmma4(v2f a, v2f b, v8f c) {
  return __builtin_amdgcn_wmma_f32_16x16x4_f32(false, a, false, b, (short)0, c, false, false);
}

// CDNA5 async global->LDS copy (ASYNCcnt-tracked). dst = LDS byte address (VGPR),
// src = 64-bit global address (VGPR pair). VGLOBAL encoding: VDST carries LDS addr.
__device__ __forceinline__ void async_b128(uint32_t lds_addr, const float* g) {
  asm volatile("global_load_async_to_lds_b128 %0, %1, off" :: "v"(lds_addr), "v"(g) : "memory");
}
__device__ __forceinline__ void wait_async0() {
  asm volatile("s_wait_asynccnt 0x0" ::: "memory");
}

// Double-buffered async-copy WMMA GEMM: out[M,Nd] = act(A[M,Kdim] @ W[Kdim,Nd] + bias)
// Block: 256 threads / 8 waves; tile 128x64; each wave computes 16x64 (4 accumulators).
template<int RELU>
__global__ __launch_bounds__(256)
void k_gemm_async(const float* __restrict__ A, const float* __restrict__ W,
                  const float* __restrict__ bias, float* __restrict__ out,
                  int Kdim, int Nd) {
  __shared__ float sA[2][MBLK_ * LDA_];   // 2 * 128*36*4 = 36864 B
  __shared__ float sB[2][KB_ * LDB_];     // 2 *  32*80*4 = 20480 B

  const int tid  = threadIdx.x;
  const int wave = tid >> 5, lane = tid & 31;
  const int half = lane >> 4, l = lane & 15;
  const int tile_m = blockIdx.x * MBLK_;
  const int tile_n = blockIdx.y * NBLK_;

  // Staging decomposition: A panel = 128x32 floats = 1024 b128 chunks (4/thread),
  //                        B panel =  32x64 floats =  512 b128 chunks (2/thread).
  const int aRow = tid >> 3, aC4 = (tid & 7) << 2;
  const int bRow = tid >> 4, bC4 = (tid & 15) << 2;
  const uint32_t ldsA = (uint32_t)(uintptr_t)(&sA[0][0]);  // LDS aperture: low 32 bits = LDS addr
  const uint32_t ldsB = (uint32_t)(uintptr_t)(&sB[0][0]);

  auto stage = [&](int buf, int k0) {
    const uint32_t aBase = ldsA + (uint32_t)(buf * MBLK_ * LDA_ * 4);
    const uint32_t bBase = ldsB + (uint32_t)(buf * KB_ * LDB_ * 4);
#pragma unroll
    for (int i = 0; i < 4; ++i) {
      const int row = aRow + i * 32;
      async_b128(aBase + (uint32_t)((row * LDA_ + aC4) * 4),
                 A + (size_t)(tile_m + row) * Kdim + k0 + aC4);
    }
#pragma unroll
    for (int i = 0; i < 2; ++i) {
      const int row = bRow + i * 16;
      async_b128(bBase + (uint32_t)((row * LDB_ + bC4) * 4),
                 W + (size_t)(k0 + row) * Nd + tile_n + bC4);
    }
  };

  v8f acc[4] = {};
  const int aOff = (wave * 16 + l) * LDA_ + 2 * half;
  const int nPanels = Kdim / KB_;

  stage(0, 0);
  wait_async0();
  __syncthreads();

  for (int p = 0; p < nPanels; ++p) {
    const int buf = p & 1;
    if (p + 1 < nPanels) stage(buf ^ 1, (p + 1) * KB_);   // overlap copy with WMMA

    const float* sa = &sA[buf][aOff];
    const float* sb = &sB[buf][l];
#pragma unroll
    for (int kl = 0; kl < KB_; kl += 4) {
      v2f a = *(const v2f*)(sa + kl);                      // A frag: VGPR0=K+2h, VGPR1=K+2h+1
      const float* br = sb + (kl + 2 * half) * LDB_;
#pragma unroll
      for (int j = 0; j < 4; ++j) {                        // 4 N-tiles reuse one A frag
        v2f b;
        b.x = br[j * 16];
        b.y = br[LDB_ + j * 16];
        acc[j] = wmma4(a, b, acc[j]);
      }
    }
    wait_async0();      // next panel landed (copies ran behind the WMMAs)
    __syncthreads();    // all waves done reading this buffer before it is re-staged
  }

#pragma unroll
  for (int j = 0; j < 4; ++j) {
    const int col = tile_n + j * 16 + l;
    const float bv = bias[col];
#pragma unroll
    for (int r = 0; r < 8; ++r) {                          // C/D: VGPR r -> row r+8*half, col l
      float v = acc[j][r] + bv;
      if (RELU) v = fmaxf(v, 0.0f);
      out[(size_t)(tile_m + wave * 16 + r + half * 8) * Nd + col] = v;
    }
  }
}

// Row-wise L2 normalization: one wave per row, wave32 shuffle reduction.
__global__ void k_normalize(float* __restrict__ e) {
  const int row  = blockIdx.x * (blockDim.x >> 5) + (threadIdx.x >> 5);
  const int lane = threadIdx.x & 31;
  float* r = e + (size_t)row * D_;
  float ss = 0.0f;
  for (int d = lane; d < D_; d += 32) { float v = r[d]; ss += v * v; }
#pragma unroll
  for (int off = 16; off > 0; off >>= 1) ss += __shfl_xor(ss, off, 32);
  const float inv = rsqrtf(ss);
  for (int d = lane; d < D_; d += 32) r[d] *= inv;
}

// Per-class embedding sums, stored transposed+padded: St[d*CP_ + c] = sum_{j:y_j=c} e[j][d]
__global__ void k_class_sums(const float* __restrict__ e, const int* __restrict__ y,
                             float* __restrict__ St) {
  const int b = blockIdx.x;
  const int d = threadIdx.x;          // 256 threads = D_
  const int c = y[b];
  const float* base = e + (size_t)b * TQ_ * D_ + d;
  float s = 0.0f;
#pragma unroll 4
  for (int t = 0; t < TQ_; ++t) s += base[(size_t)t * D_];
  atomicAdd(&St[(size_t)d * CP_ + c], s);   // global_atomic_add_f32
}

// Per-batch: scorer layer-2, softmax p, gumbel softmax, rank-based top-K gate.
__global__ void k_score_gate(const float* __restrict__ h, const float* __restrict__ w2,
                             const float* __restrict__ b2, const float* __restrict__ gumbel,
                             const int* __restrict__ y, float* __restrict__ p_out,
                             float* __restrict__ ypatch_out, float* __restrict__ wgate) {
  __shared__ float sel_s[TQ_];
  __shared__ float ys_s[TQ_];
  const int b = blockIdx.x;
  const int t = threadIdx.x;          // 64 threads = TQ_
  const int row = b * TQ_ + t;

  const float* hrow = h + (size_t)row * H_;
  float acc = b2[0];
#pragma unroll 4
  for (int j = 0; j < H_; ++j) acc += hrow[j] * w2[j];
  sel_s[t] = acc;
  __syncthreads();

  float m = -3.4e38f;
  for (int j = 0; j < TQ_; ++j) m = fmaxf(m, sel_s[j]);
  float sum = 0.0f;
  for (int j = 0; j < TQ_; ++j) sum += __expf(sel_s[j] - m);
  p_out[row] = __expf(acc - m) / sum;

  const float g = (acc + gumbel[row]) / TAU_;
  __syncthreads();
  ys_s[t] = g;
  __syncthreads();
  float m2 = -3.4e38f;
  for (int j = 0; j < TQ_; ++j) m2 = fmaxf(m2, ys_s[j]);
  float s2 = 0.0f;
  for (int j = 0; j < TQ_; ++j) s2 += __expf(ys_s[j] - m2);
  const float ys = __expf(g - m2) / s2;
  __syncthreads();
  ys_s[t] = ys;
  __syncthreads();

  int rank = 0;                        // top-K rank; ties -> lower index (matches lax.top_k)
  for (int j = 0; j < TQ_; ++j) {
    const float vj = ys_s[j];
    if (vj > ys || (vj == ys && j < t)) ++rank;
  }
  wgate[row] = (rank < KSEL_) ? 1.0f : 0.0f;
  ypatch_out[row] = (float)y[b];
}

// Final logits: (e @ St)/TEMP, -1e9 at own class, times gate. WMMA over CP_=112 padded cols.
__global__ void k_logits(const float* __restrict__ e, const float* __restrict__ St,
                         const int* __restrict__ y, const float* __restrict__ wgate,
                         float* __restrict__ out) {
  const int wave = threadIdx.x >> 5;
  const int lane = threadIdx.x & 31;
  const int half = lane >> 4;
  const int l    = lane & 15;
  const int tile_m = (blockIdx.x * 4 + wave) * 16;
  const int tile_n = blockIdx.y * 16;

  const float* arow = e + (size_t)(tile_m + l) * D_ + 2 * half;
  const float* wcol = St + tile_n + l;

  v8f acc = {};
  for (int k = 0; k < D_; k += 4) {
    v2f a = *(const v2f*)(arow + k);
    v2f b;
    b.x = wcol[(size_t)(k + 2 * half + 0) * CP_];
    b.y = wcol[(size_t)(k + 2 * half + 1) * CP_];
    acc = wmma4(a, b, acc);
  }
  const int col = tile_n + l;
  if (col < C_) {
#pragma unroll
    for (int r = 0; r < 8; ++r) {
      const int row = tile_m + r + half * 8;
      float v = acc[r] * (1.0f / TEMP_);
      if (y[row / TQ_] == col) v -= 1e9f;   // diagonal mask folds into own-class column
      v *= wgate[row];
      out[(size_t)row * C_ + col] = v;
    }
  }
}

extern "C" void kernel_launch(void* const* d_in, const int* in_sizes, int n_in,
                              void* d_out, int out_size, void* d_ws, size_t ws_size,
                              hipStream_t stream) {
  const float* feats  = (const float*)d_in[0];
  const float* proj_w = (const float*)d_in[1];
  const float* proj_b = (const float*)d_in[2];
  const float* s_w1   = (const float*)d_in[3];
  const float* s_b1   = (const float*)d_in[4];
  const float* s_w2   = (const float*)d_in[5];
  const float* s_b2   = (const float*)d_in[6];
  const float* gumbel = (const float*)d_in[7];
  const int*   y      = (const int*)d_in[8];

  // Output layout: logits [N,C] | y_patches [N] | p [B,TQ]
  float* out_logits = (float*)d_out;
  float* out_ypatch = out_logits + (size_t)N_ * C_;
  float* out_p      = out_ypatch + N_;

  // Workspace layout (~12.7 MB)
  float* embeds = (float*)d_ws;                  // N*D
  float* hbuf   = embeds + (size_t)N_ * D_;      // N*H
  float* St     = hbuf   + (size_t)N_ * H_;      // D*CP
  float* wgate  = St     + (size_t)D_ * CP_;     // N

  hipMemsetAsync(St, 0, sizeof(float) * D_ * CP_, stream);

  const dim3 blk(256);  // 8 wave32 per block
  // 1) projection GEMM (async-LDS double-buffered fp32 WMMA)
  k_gemm_async<0><<<dim3(N_ / MBLK_, D_ / NBLK_), blk, 0, stream>>>(feats, proj_w, proj_b, embeds, F_, D_);
  // 2) L2 row normalization
  k_normalize<<<dim3(N_ / 8), dim3(256), 0, stream>>>(embeds);
  // 3) per-class sums (transposed, padded)
  k_class_sums<<<dim3(B_), dim3(D_), 0, stream>>>(embeds, y, St);
  // 4) scorer layer-1 (async-LDS fp32 WMMA + relu)
  k_gemm_async<1><<<dim3(N_ / MBLK_, H_ / NBLK_), blk, 0, stream>>>(embeds, s_w1, s_b1, hbuf, D_, H_);
  // 5) scorer layer-2 + softmaxes + top-K gate + y_patches
  k_score_gate<<<dim3(B_), dim3(TQ_), 0, stream>>>(hbuf, s_w2, s_b2, gumbel, y, out_p, out_ypatch, wgate);
  // 6) gated class logits (fp32 WMMA against class sums)
  k_logits<<<dim3(N_ / 64, CP_ / 16), dim3(128), 0, stream>>>(embeds, St, y, wgate, out_logits);
}